// RoutingLMHead_9869834846661
// MI455X (gfx1250) — compile-verified
//
#include <hip/hip_runtime.h>
#include <hip/hip_bf16.h>

typedef float v2f __attribute__((ext_vector_type(2)));
typedef float v8f __attribute__((ext_vector_type(8)));

#define GEMM_TPB 256   // 8 waves
#define TOPK_TPB 256

// ---------------------------------------------------------------------------
// Kernel 1: scores[n, v] = sum_d hidden[n, d] * weight[v, d]   (exact fp32)
// V_WMMA_F32_16X16X4_F32. Each wave owns a 32(M) x 64(N) strip:
//   2 A fragments x 4 B fragments -> 8 WMMAs per K-step, 6 loads (b64 each).
// Register double-buffer: prefetch K+4 fragments before issuing K's WMMAs so
// the wave overlaps global loads with matrix math instead of waiting to 0.
// Block = 8 waves (4 row-groups x 2 col-groups) = 128 rows x 128 cols.
// grid.x sweeps V so co-resident blocks share weight tiles through L2.
// ---------------------------------------------------------------------------
__global__ __launch_bounds__(GEMM_TPB)
void gemm_scores_kernel(const float* __restrict__ hidden,
                        const float* __restrict__ weight,
                        float* __restrict__ out,
                        int N, int V, int D) {
    const int lane = threadIdx.x & 31;
    const int wave = threadIdx.x >> 5;
    const int lm   = lane & 15;      // M (A) / N (B, C)
    const int half = lane >> 4;      // selects K pair 0/1 vs 2/3
    const int wrow = wave >> 1;      // 0..3
    const int wcol = wave & 1;       // 0..1

    const int rowBase = blockIdx.y * 128 + wrow * 32;
    const int colBase = blockIdx.x * 128 + wcol * 64;

    const float* aptr[2];
#pragma unroll
    for (int r = 0; r < 2; ++r) {
        int ar = rowBase + r * 16 + lm;
        if (ar > N - 1) ar = N - 1;                    // clamp row tail
        aptr[r] = hidden + (size_t)ar * D + 2 * half;
    }
    const float* bptr[4];
#pragma unroll
    for (int t = 0; t < 4; ++t) {
        int c = colBase + t * 16 + lm;
        if (c > V - 1) c = V - 1;                      // clamp col tail
        bptr[t] = weight + (size_t)c * D + 2 * half;
    }

    v8f acc[2][4];
#pragma unroll
    for (int r = 0; r < 2; ++r)
#pragma unroll
        for (int t = 0; t < 4; ++t) acc[r][t] = v8f{0};

    // prologue: fragments for k = 0
    v2f a[2], b[4];
#pragma unroll
    for (int r = 0; r < 2; ++r) a[r] = *(const v2f*)(aptr[r]);
#pragma unroll
    for (int t = 0; t < 4; ++t) b[t] = *(const v2f*)(bptr[t]);

#pragma unroll 2
    for (int k = 0; k < D; k += 4) {
        const int kn = (k + 4 < D) ? (k + 4) : k;      // last iter: harmless reload
        v2f an[2], bn[4];
#pragma unroll
        for (int r = 0; r < 2; ++r) an[r] = *(const v2f*)(aptr[r] + kn);
#pragma unroll
        for (int t = 0; t < 4; ++t) bn[t] = *(const v2f*)(bptr[t] + kn);

#pragma unroll
        for (int r = 0; r < 2; ++r)
#pragma unroll
            for (int t = 0; t < 4; ++t)
                acc[r][t] = __builtin_amdgcn_wmma_f32_16x16x4_f32(
                    false, a[r], false, b[t], (short)0, acc[r][t], false, false);

#pragma unroll
        for (int r = 0; r < 2; ++r) a[r] = an[r];
#pragma unroll
        for (int t = 0; t < 4; ++t) b[t] = bn[t];
    }

    // C/D layout: VGPR j -> lanes 0-15: (M=j, N=lane), lanes 16-31: (M=j+8, N=lane-16)
#pragma unroll
    for (int r = 0; r < 2; ++r) {
#pragma unroll
        for (int t = 0; t < 4; ++t) {
            const int col = colBase + t * 16 + lm;
            if (col < V) {
#pragma unroll
                for (int j = 0; j < 8; ++j) {
                    const int row = rowBase + r * 16 + j + 8 * half;
                    if (row < N) out[(size_t)row * V + col] = acc[r][t][j];
                }
            }
        }
    }
}

// ---------------------------------------------------------------------------
// Kernel 2: in-place top-k mask per row. Exact k-th-largest via 4x 8-bit
// radix-select on the order-preserving uint mapping of fp32, then rewrite:
// keep key > T; keep key == T for the first `rem` by index (deterministic,
// matches top_k index-order tie-break); everything else := -inf.
// One block per row; row (~200KB) stays L2-resident across the 6 passes.
// ---------------------------------------------------------------------------
__device__ __forceinline__ unsigned f2ord(float f) {
    unsigned b = __float_as_uint(f);
    return (b & 0x80000000u) ? ~b : (b | 0x80000000u);
}

__global__ __launch_bounds__(TOPK_TPB)
void topk_mask_kernel(float* __restrict__ scores, const int* __restrict__ topk_ptr, int V) {
    const int tid = threadIdx.x;
    float* rowp = scores + (size_t)blockIdx.x * V;
    const int k = topk_ptr[0];

    __shared__ unsigned bins[256];
    __shared__ unsigned eqcnt[TOPK_TPB];
    __shared__ unsigned s_prefix;
    __shared__ int s_rem;

    unsigned prefix = 0;
    int rem = k;

    // 4 radix rounds, high byte first
    for (int r = 0; r < 4; ++r) {
        for (int b = tid; b < 256; b += TOPK_TPB) bins[b] = 0u;
        __syncthreads();

        const int shift = 24 - 8 * r;
        for (int i = tid; i < V; i += TOPK_TPB) {
            unsigned key = f2ord(rowp[i]);
            bool part = (r == 0) || ((key >> (shift + 8)) == prefix);
            if (part) atomicAdd(&bins[(key >> shift) & 255u], 1u);
        }
        __syncthreads();

        if (tid == 0) {
            unsigned cum = 0;
            int d = 255;
            for (; d > 0; --d) {
                cum += bins[d];
                if ((int)cum >= rem) break;
            }
            if ((int)cum < rem) cum += bins[0];        // d==0 fallthrough
            s_prefix = (prefix << 8) | (unsigned)d;
            s_rem    = rem - (int)(cum - bins[d]);     // strictly-greater keys locked in
        }
        __syncthreads();
        prefix = s_prefix;
        rem    = s_rem;
        __syncthreads();
    }

    const unsigned T = prefix;   // exact k-th largest key
    // rem = number of keys == T that belong in the top-k (>=1)

    // Tie-rank pass over contiguous chunks so equal-key rank follows index order.
    const int chunk = (V + TOPK_TPB - 1) / TOPK_TPB;
    const int beg = tid * chunk;
    const int end = (beg + chunk < V) ? (beg + chunk) : V;

    unsigned myeq = 0;
    for (int i = beg; i < end; ++i)
        if (f2ord(rowp[i]) == T) ++myeq;
    eqcnt[tid] = myeq;
    __syncthreads();

    if (tid == 0) {              // serial exclusive prefix over 256 entries
        unsigned run = 0;
        for (int t = 0; t < TOPK_TPB; ++t) {
            unsigned c = eqcnt[t];
            eqcnt[t] = run;
            run += c;
        }
    }
    __syncthreads();

    const float NEG_INF = __uint_as_float(0xFF800000u);
    unsigned eqrank = eqcnt[tid];
    for (int i = beg; i < end; ++i) {
        unsigned key = f2ord(rowp[i]);
        bool keep;
        if (key > T) {
            keep = true;
        } else if (key == T) {
            keep = ((int)eqrank < rem);
            ++eqrank;
        } else {
            keep = false;
        }
        if (!keep) rowp[i] = NEG_INF;
    }
}

// ---------------------------------------------------------------------------
extern "C" void kernel_launch(void* const* d_in, const int* in_sizes, int n_in,
                              void* d_out, int out_size, void* d_ws, size_t ws_size,
                              hipStream_t stream) {
    const float* hidden = (const float*)d_in[0];   // [B, S, D] fp32
    const float* weight = (const float*)d_in[1];   // [V, D]    fp32
    const int*   topk   = (const int*)d_in[2];     // scalar
    float* out = (float*)d_out;                    // [B, S, V] fp32

    const int D = 1024;
    const int N = in_sizes[0] / D;   // 4096
    const int V = in_sizes[1] / D;   // 50257

    dim3 g1((V + 127) / 128, (N + 127) / 128);
    gemm_scores_kernel<<<g1, GEMM_TPB, 0, stream>>>(hidden, weight, out, N, V, D);

    topk_mask_kernel<<<N, TOPK_TPB, 0, stream>>>(out, topk, V);
}